// InteractionNet_35931696398847
// MI455X (gfx1250) — compile-verified
//
#include <hip/hip_runtime.h>
#include <hip/hip_bf16.h>

typedef __attribute__((ext_vector_type(2))) float v2f;
typedef __attribute__((ext_vector_type(4))) float v4f;
typedef __attribute__((ext_vector_type(8))) float v8f;

#define FDIM 128
#define GDIM 64
#define NPLANES 3

// ------------------------------------------------------------------
// Workspace layout (floats):
//   [0,                3*G*F)  : per-plane, per-graph weighted sums
//   [3*G*F, 3*G*F+3*G)         : per-plane, per-graph counts
// ------------------------------------------------------------------

__global__ void init_ws_kernel(float* __restrict__ ws, int n) {
    int i = blockIdx.x * blockDim.x + threadIdx.x;
    if (i < n) ws[i] = 0.0f;
}

// Streaming attention pool: one plane per blockIdx.y; one wave32 per row per
// iteration (lane loads 4 floats -> 32*4 = 128 features). Wave-reduce the
// attention dot, sigmoid, scatter a*x into a 32KB LDS accumulator with
// ds_add_f32 atomics, then flush once per block with global fp32 atomics.
// x is a read-once 307MB stream (> 192MB L2) -> non-temporal loads.
__global__ __launch_bounds__(256) void pool_kernel(
    const float* __restrict__ x_u, const float* __restrict__ x_v,
    const float* __restrict__ x_y,
    const int* __restrict__ idx_u, const int* __restrict__ idx_v,
    const int* __restrict__ idx_y,
    const float* __restrict__ w_u, const float* __restrict__ b_u,
    const float* __restrict__ w_v, const float* __restrict__ b_v,
    const float* __restrict__ w_y, const float* __restrict__ b_y,
    float* __restrict__ g_sums,   // [3][G][F]
    float* __restrict__ g_counts, // [3][G]
    int n)
{
    __shared__ float s_sum[GDIM * FDIM];
    __shared__ float s_cnt[GDIM];

    const int plane = blockIdx.y;
    const float* x; const int* idx; const float* w; float bias;
    if (plane == 0)      { x = x_u; idx = idx_u; w = w_u; bias = b_u[0]; }
    else if (plane == 1) { x = x_v; idx = idx_v; w = w_v; bias = b_v[0]; }
    else                 { x = x_y; idx = idx_y; w = w_y; bias = b_y[0]; }

    for (int i = threadIdx.x; i < GDIM * FDIM; i += blockDim.x) s_sum[i] = 0.0f;
    if (threadIdx.x < GDIM) s_cnt[threadIdx.x] = 0.0f;
    __syncthreads();

    const int lane = threadIdx.x & 31;
    const int wave = threadIdx.x >> 5;
    const int wavesPerBlock = blockDim.x >> 5;
    const int waveGlobal = blockIdx.x * wavesPerBlock + wave;
    const int waveStride = gridDim.x * wavesPerBlock;

    const v4f wv = ((const v4f*)w)[lane];   // w_att is [F,1] contiguous

    for (int r = waveGlobal; r < n; r += waveStride) {
        const v4f xv =
            __builtin_nontemporal_load(((const v4f*)(x + (size_t)r * FDIM)) + lane);
        float t = xv.x * wv.x + xv.y * wv.y + xv.z * wv.z + xv.w * wv.w;
        // wave32 all-reduce
        #pragma unroll
        for (int m = 16; m > 0; m >>= 1) t += __shfl_xor(t, m, 32);
        const float a = 1.0f / (1.0f + __expf(-(t + bias)));
        const int g = idx[r];
        float* dst = &s_sum[g * FDIM + lane * 4];
        atomicAdd(dst + 0, a * xv.x);
        atomicAdd(dst + 1, a * xv.y);
        atomicAdd(dst + 2, a * xv.z);
        atomicAdd(dst + 3, a * xv.w);
        if (lane == 0) atomicAdd(&s_cnt[g], 1.0f);
    }
    __syncthreads();

    float* gsum = g_sums + (size_t)plane * GDIM * FDIM;
    for (int i = threadIdx.x; i < GDIM * FDIM; i += blockDim.x)
        atomicAdd(&gsum[i], s_sum[i]);
    if (threadIdx.x < GDIM)
        atomicAdd(&g_counts[plane * GDIM + threadIdx.x], s_cnt[threadIdx.x]);
}

// Final tiny GEMM on the fp32 matrix pipe:
//   E[64][384] (normalized pooled features) @ w_net[384][128] + b_net
// One wave per 16x16 output tile; 4x8 = 32 tiles = 32 waves total.
// Uses V_WMMA_F32_16X16X4_F32 with K stepped by 4 (96 MACs per tile).
// Plane-outer loop hoists the 1/max(count,1) reciprocal to 3 divisions
// per wave (instead of one per K-step).
// A 16x4 layout: lane = M (mod 16), half-wave selects K pair {0,1}/{2,3}.
// B 4x16 layout mirrors A. C/D: VGPR v -> M = v + 8*(lane>=16), N = lane%16.
__global__ __launch_bounds__(128) void gemm_wmma_kernel(
    const float* __restrict__ g_sums,   // [3][G][F]
    const float* __restrict__ g_counts, // [3][G]
    const float* __restrict__ w_net,    // [384][128]
    const float* __restrict__ b_net,    // [128]
    float* __restrict__ out)            // [64][128]
{
    const int lane = threadIdx.x & 31;
    const int wave = threadIdx.x >> 5;
    const int tile = blockIdx.x * (blockDim.x >> 5) + wave; // 0..31
    const int tileM = tile >> 3;   // 0..3
    const int tileN = tile & 7;    // 0..7

    const int half = lane >> 4;    // 0 or 1
    const int l16  = lane & 15;

    const int n = tileN * 16 + l16;   // output column for B/C/D fragments
    const int m = tileM * 16 + l16;   // A-fragment row

    v8f c;
    const float bias = b_net[n];
    #pragma unroll
    for (int i = 0; i < 8; ++i) c[i] = bias;

    for (int plane = 0; plane < NPLANES; ++plane) {
        const float cnt = g_counts[plane * GDIM + m];
        const float inv = 1.0f / fmaxf(cnt, 1.0f);          // hoisted per plane
        const float* sp = g_sums + ((size_t)plane * GDIM + m) * FDIM;
        const float* wp = w_net + (size_t)plane * FDIM * FDIM;

        #pragma unroll 8
        for (int k4 = 0; k4 < FDIM; k4 += 4) {
            const int k0 = k4 + half * 2;     // this half-wave's K pair (even)
            // A fragment: E[m][k0..k0+1] = sums * inv (8B-aligned paired load)
            v2f a = *(const v2f*)(sp + k0);
            a.x *= inv;
            a.y *= inv;
            // B fragment: w_net rows k0, k0+1 at column n (coalesced)
            v2f b;
            b.x = wp[(size_t)k0 * FDIM + n];
            b.y = wp[(size_t)(k0 + 1) * FDIM + n];
            c = __builtin_amdgcn_wmma_f32_16x16x4_f32(
                    /*neg_a=*/false, a, /*neg_b=*/false, b,
                    /*c_mod=*/(short)0, c, /*reuse_a=*/false, /*reuse_b=*/false);
        }
    }

    #pragma unroll
    for (int v = 0; v < 8; ++v) {
        out[(size_t)(tileM * 16 + v + 8 * half) * FDIM + n] = c[v];
    }
}

extern "C" void kernel_launch(void* const* d_in, const int* in_sizes, int n_in,
                              void* d_out, int out_size, void* d_ws, size_t ws_size,
                              hipStream_t stream) {
    // Input order from setup_inputs():
    // 0:x_u 1:x_v 2:x_y 3:idx_u 4:idx_v 5:idx_y 6:num_graphs
    // 7:w_att_u 8:b_att_u 9:w_att_v 10:b_att_v 11:w_att_y 12:b_att_y
    // 13:w_net 14:b_net
    const float* x_u = (const float*)d_in[0];
    const float* x_v = (const float*)d_in[1];
    const float* x_y = (const float*)d_in[2];
    const int* idx_u = (const int*)d_in[3];
    const int* idx_v = (const int*)d_in[4];
    const int* idx_y = (const int*)d_in[5];
    const float* w_u = (const float*)d_in[7];
    const float* b_u = (const float*)d_in[8];
    const float* w_v = (const float*)d_in[9];
    const float* b_v = (const float*)d_in[10];
    const float* w_y = (const float*)d_in[11];
    const float* b_y = (const float*)d_in[12];
    const float* w_net = (const float*)d_in[13];
    const float* b_net = (const float*)d_in[14];
    float* out = (float*)d_out;

    const int n = in_sizes[0] / FDIM;   // hits per plane (200000)

    float* g_sums = (float*)d_ws;                       // 3*G*F floats
    float* g_counts = g_sums + NPLANES * GDIM * FDIM;   // 3*G floats

    const int ws_elems = NPLANES * GDIM * FDIM + NPLANES * GDIM;
    init_ws_kernel<<<(ws_elems + 255) / 256, 256, 0, stream>>>(g_sums, ws_elems);

    dim3 poolGrid(256, NPLANES);
    pool_kernel<<<poolGrid, 256, 0, stream>>>(
        x_u, x_v, x_y, idx_u, idx_v, idx_y,
        w_u, b_u, w_v, b_v, w_y, b_y,
        g_sums, g_counts, n);

    // 32 tiles -> 8 blocks x 4 waves (128 threads)
    gemm_wmma_kernel<<<8, 128, 0, stream>>>(g_sums, g_counts, w_net, b_net, out);

    (void)in_sizes; (void)n_in; (void)out_size; (void)ws_size;
}